// MoELayer_84954453115232
// MI455X (gfx1250) — compile-verified
//
#include <hip/hip_runtime.h>
#include <math.h>

// MoE with top-2 gating. The reference's dense all-expert einsum (~137 GFLOP)
// is dead except features o=0,1 of the selected experts, so everything
// collapses to a [tokens x 1024] x [1024 x 32] GEMM (8 gate cols + 16
// candidate cols + 8 don't-care pad cols) + scalar epilogue + broadcast store.
// Memory-bound: ~64MB traffic -> ~2.8us at 23.3 TB/s.
//
// Round 2 changes:
//  - pad columns 24..31 are never read by the epilogue -> no zero-masking,
//    removes a v_pk_mul_f32 per WMMA pair from the hot loop.
//  - K split 4 ways across 4 waves per workgroup (2048 waves total) for
//    latency hiding; partial 16x32 tiles reduced through LDS.

typedef __attribute__((ext_vector_type(2))) float v2f;
typedef __attribute__((ext_vector_type(4))) float v4f;
typedef __attribute__((ext_vector_type(8))) float v8f;

#define MOE_B    4
#define MOE_S    2048
#define MOE_DIN  1024
#define MOE_DOUT 1024
#define MOE_E    8
#define MOE_K    2
#define KSPLIT   4                      // waves per tile (K sliced 4 ways)
#define KSLICE   (MOE_DIN / KSPLIT)     // 256 floats per wave

__global__ __launch_bounds__(32 * KSPLIT)
void moe_wmma_kernel(const float* __restrict__ x,             // [B*S, DIN]
                     const float* __restrict__ gate_w,        // [E, DIN]
                     const float* __restrict__ gate_b,        // [E]
                     const float* __restrict__ expert_biases, // [E]
                     const float* __restrict__ expert_w,      // [E, DOUT, DIN]
                     const float* __restrict__ expert_b,      // [E, DOUT]
                     float* __restrict__ out,                 // [B*S, DOUT]
                     int* __restrict__ idx_out)               // [B*S, K]
{
    __shared__ float P[KSPLIT][16 * 32];   // partial product tiles
    __shared__ float Wt[16];               // weighted scalar per token

    const int tid  = threadIdx.x;          // 0..127
    const int lane = tid & 31;             // wave32 lane
    const int wv   = tid >> 5;             // K-slice id 0..3
    const int lrow = lane & 15;
    const int half = lane >> 4;            // 0: K=k,k+1   1: K=k+2,k+3
    const int m0   = blockIdx.x * 16;      // token tile base

    // ---- A fragment source: row m0+lrow of x, this wave's K slice ----
    const float* aPtr = x + (size_t)(m0 + lrow) * MOE_DIN + wv * KSLICE;

    // ---- B tile0 column (n = lrow): 0..7 gate rows, 8..15 expert_w[e][o=0] ----
    const float* b0Ptr = ((lrow < MOE_E)
        ? (gate_w + (size_t)lrow * MOE_DIN)
        : (expert_w + (size_t)(lrow - MOE_E) * MOE_DOUT * MOE_DIN)) + wv * KSLICE;
    // ---- B tile1 column (n = 16+lrow): 16..23 expert_w[e][o=1];
    //      24..31 alias experts 0..7 again -> garbage, never read ----
    const float* b1Ptr = expert_w + (size_t)(lrow & 7) * MOE_DOUT * MOE_DIN
                         + MOE_DIN + wv * KSLICE;

    v8f acc0 = {};   // cols 0..15
    v8f acc1 = {};   // cols 16..31
    const int kBase = half * 2;

#pragma unroll 8
    for (int kk = 0; kk < KSLICE / 4; ++kk) {     // 64 k-steps of 4
        const int ko = kk * 4 + kBase;
        v2f a  = *(const v2f*)(aPtr  + ko);
        v2f b0 = *(const v2f*)(b0Ptr + ko);
        v2f b1 = *(const v2f*)(b1Ptr + ko);
        acc0 = __builtin_amdgcn_wmma_f32_16x16x4_f32(false, a, false, b0,
                                                     (short)0, acc0, false, false);
        acc1 = __builtin_amdgcn_wmma_f32_16x16x4_f32(false, a, false, b1,
                                                     (short)0, acc1, false, false);
    }

    // ---- spill partial D tiles: lanes 0-15 hold M=r, lanes 16-31 hold M=8+r ----
#pragma unroll
    for (int r = 0; r < 8; ++r) {
        const int M = half ? (8 + r) : r;
        P[wv][M * 32 + lrow]      = acc0[r];
        P[wv][M * 32 + 16 + lrow] = acc1[r];
    }
    __syncthreads();

    // ---- per-token epilogue (threads 0..15): reduce K slices, sigmoid gate,
    //      top-2 (stable: lower index wins ties), normalize, combine ----
    if (tid < 16) {
        float row[24];
#pragma unroll
        for (int c = 0; c < 24; ++c) {
            row[c] = P[0][tid * 32 + c] + P[1][tid * 32 + c]
                   + P[2][tid * 32 + c] + P[3][tid * 32 + c];
        }
        float probs[MOE_E];
#pragma unroll
        for (int e = 0; e < MOE_E; ++e) {
            const float l = row[e] + gate_b[e] + expert_biases[e];
            probs[e] = 1.0f / (1.0f + expf(-l));
        }
        int i0 = 0; float p0 = probs[0];
#pragma unroll
        for (int e = 1; e < MOE_E; ++e)
            if (probs[e] > p0) { p0 = probs[e]; i0 = e; }
        int i1 = 0; float p1 = -1.0f;
#pragma unroll
        for (int e = 0; e < MOE_E; ++e)
            if (e != i0 && probs[e] > p1) { p1 = probs[e]; i1 = e; }

        const float inv  = 1.0f / (p0 + p1);
        const float val0 = row[8  + i0] + expert_b[(size_t)i0 * MOE_DOUT + 0];
        const float val1 = row[16 + i1] + expert_b[(size_t)i1 * MOE_DOUT + 1];
        Wt[tid] = val0 * (p0 * inv) + val1 * (p1 * inv);

        const int tok = m0 + tid;
        idx_out[tok * MOE_K + 0] = i0;
        idx_out[tok * MOE_K + 1] = i1;
    }
    __syncthreads();

    // ---- coalesced broadcast store: out[m0+r][:] = Wt[r], b128 per thread ----
    v4f* out4 = (v4f*)(out + (size_t)m0 * MOE_DOUT);
#pragma unroll
    for (int r = 0; r < 16; ++r) {
        const float v = Wt[r];
        const v4f q = {v, v, v, v};
#pragma unroll
        for (int it = 0; it < MOE_DOUT / 4 / (32 * KSPLIT); ++it) {   // 2 iters
            out4[r * (MOE_DOUT / 4) + it * (32 * KSPLIT) + tid] = q;
        }
    }
}

extern "C" void kernel_launch(void* const* d_in, const int* in_sizes, int n_in,
                              void* d_out, int out_size, void* d_ws, size_t ws_size,
                              hipStream_t stream) {
    const float* x             = (const float*)d_in[0];
    const float* gate_w        = (const float*)d_in[1];
    const float* gate_b        = (const float*)d_in[2];
    const float* expert_biases = (const float*)d_in[3];
    const float* expert_w      = (const float*)d_in[4];
    const float* expert_b      = (const float*)d_in[5];

    float* out = (float*)d_out;
    // tuple output: [B*S*DOUT] f32 final_output, then [B*S*K] i32 top_k_indices
    int* idx_out = (int*)(out + (size_t)MOE_B * MOE_S * MOE_DOUT);

    const int tiles = (MOE_B * MOE_S) / 16;  // 512 tiles, 4 waves each
    moe_wmma_kernel<<<tiles, 32 * KSPLIT, 0, stream>>>(
        x, gate_w, gate_b, expert_biases, expert_w, expert_b, out, idx_out);

    (void)in_sizes; (void)n_in; (void)out_size; (void)d_ws; (void)ws_size;
}